// DCNv4_84104049590502
// MI455X (gfx1250) — compile-verified
//
#include <hip/hip_runtime.h>
#include <hip/hip_bf16.h>

// ---------------------------------------------------------------------------
// DCNv4 forward for MI455X (gfx1250), wave32, WMMA f32 16x16x4.
//
// Pipeline:
//   x  = input @ value_w^T  + value_b          (WMMA GEMM, full 65536 rows)
//   per slab of 4096 rows:
//     om = input @ offmask_w^T + offmask_b     (WMMA GEMM)
//     y  = dcnv4_core(x, om)                   (gather / trilinear; om via LDS)
//     out= y @ out_w^T + out_b                 (WMMA GEMM, direct to d_out)
// ---------------------------------------------------------------------------

typedef __attribute__((ext_vector_type(2))) float v2f;
typedef __attribute__((ext_vector_type(8))) float v8f;
typedef __attribute__((ext_vector_type(4))) int   v4i;

#define NN     2
#define DD     32
#define HH     32
#define WW     32
#define CC     128
#define LL     (DD * HH * WW)          // 32768
#define GG     8
#define CG     16
#define KK     27
#define OMD    864                     // G * K * 4
#define KDIM   128                     // GEMM reduction dim (== CC)
#define SLAB   4096                    // rows per pipeline slab (divides L)

#if defined(__has_builtin)
#if __has_builtin(__builtin_amdgcn_global_load_async_to_lds_b128)
#define USE_ASYNC_LDS 1
#endif
#endif

#define AS1 __attribute__((address_space(1)))
#define AS3 __attribute__((address_space(3)))

// ---------------------------------------------------------------------------
// GEMM:  Cout[m, n] = sum_k A[m, k] * W[n, k] + bias[n]
//   A: row-major [M, 128], W: row-major [Nc, 128], Cout: row-major [M, Nc].
// One wave computes a 32x32 tile: 2 A-frags + 2 B-frags feed 4 WMMAs per
// k-step (halves fragment traffic vs. 16x16 tiles). 128 WMMA per wave-tile.
// Grid must satisfy gridDim.x * 8 == (M/32) * (Nc/32)  (exact, EXEC all-ones).
// ---------------------------------------------------------------------------
__global__ __launch_bounds__(256)
void gemm_xwt_f32(const float* __restrict__ A,
                  const float* __restrict__ W,
                  const float* __restrict__ bias,
                  float* __restrict__ Cout,
                  int M, int Nc)
{
    const int lane = threadIdx.x & 31;
    const int wave = threadIdx.x >> 5;
    const int ntn  = Nc >> 5;                       // 32-wide tiles along N
    const int tile = blockIdx.x * 8 + wave;

    const int m0 = (tile / ntn) << 5;
    const int n0 = (tile % ntn) << 5;

    const int row = lane & 15;                      // M-row for A frags, N-col for B frags
    const int hi  = lane >> 4;                      // 0: K pair {0,1}, 1: K pair {2,3}

    const float* __restrict__ A0 = A + (size_t)(m0 + row) * KDIM + 2 * hi;
    const float* __restrict__ A1 = A0 + (size_t)16 * KDIM;
    const float* __restrict__ W0 = W + (size_t)(n0 + row) * KDIM + 2 * hi;
    const float* __restrict__ W1 = W0 + (size_t)16 * KDIM;

    v8f c00 = {}, c01 = {}, c10 = {}, c11 = {};
#pragma unroll 2
    for (int k0 = 0; k0 < KDIM; k0 += 4) {
        const v2f a0 = *(const v2f*)(A0 + k0);      // lanes 0-15: K={k0,k0+1}; 16-31: K={k0+2,k0+3}
        const v2f a1 = *(const v2f*)(A1 + k0);
        const v2f b0 = *(const v2f*)(W0 + k0);      // same K striping, col = lane&15
        const v2f b1 = *(const v2f*)(W1 + k0);
        c00 = __builtin_amdgcn_wmma_f32_16x16x4_f32(false, a0, false, b0, (short)0, c00, false, false);
        c01 = __builtin_amdgcn_wmma_f32_16x16x4_f32(false, a0, false, b1, (short)0, c01, false, false);
        c10 = __builtin_amdgcn_wmma_f32_16x16x4_f32(false, a1, false, b0, (short)0, c10, false, false);
        c11 = __builtin_amdgcn_wmma_f32_16x16x4_f32(false, a1, false, b1, (short)0, c11, false, false);
    }

    const float bn0 = bias[n0 + row];
    const float bn1 = bias[n0 + 16 + row];
#pragma unroll
    for (int i = 0; i < 8; ++i) {
        const int mA = m0 + i + 8 * hi;             // C/D layout: VGPR i -> M = i + 8*(lane>=16)
        float* r0 = Cout + (size_t)mA * Nc + n0 + row;
        float* r1 = r0 + (size_t)16 * Nc;
        r0[0]  = c00[i] + bn0;
        r0[16] = c01[i] + bn1;
        r1[0]  = c10[i] + bn0;
        r1[16] = c11[i] + bn1;
    }
}

// ---------------------------------------------------------------------------
// DCNv4 core: one thread per (row, g, c) output element; 256-thread block
// covers exactly 2 rows. Per-block om data (2 x 864 floats) is staged into
// LDS (async global->LDS on gfx1250 when available), then read broadcast.
//   x : [N*L, 128] value features (group g occupies channels g*16 .. g*16+15)
//   om: [rows, 864] slab; per group: 81 offsets (k*3+{d,h,w}) then 27 masks.
//   y : [rows, 128] slab output.
// Grid must be exact: rows * 128 == gridDim.x * 256.
// ---------------------------------------------------------------------------
__global__ __launch_bounds__(256)
void dcnv4_core(const float* __restrict__ x,
                const float* __restrict__ om,
                float* __restrict__ y,
                int rowBase)
{
    __shared__ float som[2 * OMD];                  // 6912 B

    const int tid   = threadIdx.x;
    const int idx   = blockIdx.x * 256 + tid;
    const int c128  = idx & (CC - 1);               // channel within 128
    const int g     = c128 >> 4;                    // group
    const int lrow  = idx >> 7;                     // row within slab
    const int lrow0 = (blockIdx.x * 256) >> 7;      // first row of this block

    // ---- stage om[lrow0 .. lrow0+1] into LDS ----
    const float* __restrict__ gom = om + (size_t)lrow0 * OMD;   // 3456B rows -> 16B aligned
#if defined(USE_ASYNC_LDS)
    for (int i = tid; i < (2 * OMD) / 4; i += 256) {            // 432 x b128 chunks
        __builtin_amdgcn_global_load_async_to_lds_b128(
            (AS1 v4i*)(void*)(gom + i * 4),
            (AS3 v4i*)(void*)(som + i * 4),
            /*offset=*/0, /*cpol=*/0);
    }
    asm volatile("s_wait_asynccnt 0x0" ::: "memory");
    __syncthreads();
#else
    for (int i = tid; i < 2 * OMD; i += 256) som[i] = gom[i];
    __syncthreads();
#endif

    const int row = rowBase + lrow;                 // global row in [0, N*L)
    const int n = row / LL;
    const int l = row - n * LL;
    const int d = l >> 10;
    const int h = (l >> 5) & 31;
    const int w = l & 31;

    const float* omp = som + (lrow - lrow0) * OMD + g * (KK * 4);   // LDS, broadcast reads
    const float* __restrict__ xb = x + (size_t)n * LL * CC + c128;  // index by flat*CC

    float acc = 0.0f;
#pragma unroll 3
    for (int k = 0; k < KK; ++k) {
        const int kz = k / 9;
        const int ky = (k / 3) % 3;
        const int kx = k % 3;

        const float pd = (float)(d + kz - 1) + omp[k * 3 + 0];
        const float ph = (float)(h + ky - 1) + omp[k * 3 + 1];
        const float pw = (float)(w + kx - 1) + omp[k * 3 + 2];
        const float mk = omp[81 + k];

        const float fd = floorf(pd), fh = floorf(ph), fw = floorf(pw);
        const float td = pd - fd,    th = ph - fh,    tw = pw - fw;
        const int id = (int)fd, ih = (int)fh, iw = (int)fw;

        float s = 0.0f;
#pragma unroll
        for (int cz = 0; cz < 2; ++cz) {
            const int   zd = id + cz;
            const float wd = cz ? td : 1.0f - td;
            if (zd < 0 || zd >= DD) continue;
#pragma unroll
            for (int cy = 0; cy < 2; ++cy) {
                const int   zh = ih + cy;
                const float wh = cy ? th : 1.0f - th;
                if (zh < 0 || zh >= HH) continue;
#pragma unroll
                for (int cx = 0; cx < 2; ++cx) {
                    const int   zw = iw + cx;
                    const float wc = cx ? tw : 1.0f - tw;
                    if (zw < 0 || zw >= WW) continue;
                    const int flat = (zd * HH + zh) * WW + zw;
                    s += xb[(size_t)flat * CC] * (wd * wh * wc);
                }
            }
        }
        acc += s * mk;
    }
    y[(size_t)lrow * CC + c128] = acc;
}

// ---------------------------------------------------------------------------
// Host-side orchestration (all launches on `stream`; graph-capture safe).
// ---------------------------------------------------------------------------
extern "C" void kernel_launch(void* const* d_in, const int* in_sizes, int n_in,
                              void* d_out, int out_size, void* d_ws, size_t ws_size,
                              hipStream_t stream)
{
    const float* input     = (const float*)d_in[0];   // [N, D, H, W, C] == [N*L, C]
    const float* value_w   = (const float*)d_in[1];   // [C, C]
    const float* value_b   = (const float*)d_in[2];   // [C]
    const float* offmask_w = (const float*)d_in[3];   // [864, C]
    const float* offmask_b = (const float*)d_in[4];   // [864]
    const float* out_w     = (const float*)d_in[5];   // [C, C]
    const float* out_b     = (const float*)d_in[6];   // [C]
    float*       out       = (float*)d_out;           // [N*L, C]

    const int totalRows = NN * LL;                    // 65536

    // Workspace layout:
    //   x       : totalRows * 128 floats  (33.5 MB)
    //   om_slab : SLAB * 864 floats       (14.2 MB)
    //   y_slab  : SLAB * 128 floats       ( 2.1 MB)
    float* x   = (float*)d_ws;
    float* omS = x   + (size_t)totalRows * CC;
    float* yS  = omS + (size_t)SLAB * OMD;

    // 1) Value projection over all rows:  x = input @ value_w^T + value_b
    {
        const int tiles = (totalRows >> 5) * (CC >> 5);   // 2048 * 4 = 8192
        gemm_xwt_f32<<<tiles / 8, 256, 0, stream>>>(input, value_w, value_b, x,
                                                    totalRows, CC);
    }

    // 2) Slab pipeline: offmask GEMM -> core -> out GEMM
    for (int rowBase = 0; rowBase < totalRows; rowBase += SLAB) {
        // om_slab = input[rowBase:rowBase+SLAB] @ offmask_w^T + offmask_b
        {
            const int tiles = (SLAB >> 5) * (OMD >> 5);   // 128 * 27 = 3456
            gemm_xwt_f32<<<tiles / 8, 256, 0, stream>>>(
                input + (size_t)rowBase * CC, offmask_w, offmask_b, omS,
                SLAB, OMD);
        }
        // y_slab = dcnv4_core(x, om_slab)
        {
            const int threads = SLAB * CC;                // 524288
            dcnv4_core<<<threads / 256, 256, 0, stream>>>(x, omS, yS, rowBase);
        }
        // out[rowBase:..] = y_slab @ out_w^T + out_b
        {
            const int tiles = (SLAB >> 5) * (CC >> 5);    // 128 * 4 = 512
            gemm_xwt_f32<<<tiles / 8, 256, 0, stream>>>(
                yS, out_w, out_b, out + (size_t)rowBase * CC,
                SLAB, CC);
        }
    }
}